// StructuredLSNN_2654289789590
// MI455X (gfx1250) — compile-verified
//
#include <hip/hip_runtime.h>

// ---------------------------------------------------------------------------
// StructuredLSNN forward on gfx1250 (MI455X), f32 WMMA (V_WMMA_F32_16X16X4_F32)
//
// Latency-bound sequential scan (T=4096). Per step, per block (16 batch rows):
//   state/cur computed in ONE phase from s_{t-1} via fused weights
//   (W2 = A^T @ enc_w^T, b2 = B^T @ enc_w^T), one split-barrier per step.
// Each wave owns 2 output tiles; WMMA stream is interleaved with 4
// independent accumulators (dep distance 4) to keep the XDL pipe full.
// Spike stores overlap the barrier via s_barrier_signal / s_barrier_wait.
// ---------------------------------------------------------------------------

typedef __attribute__((ext_vector_type(2))) float v2f;
typedef __attribute__((ext_vector_type(8))) float v8f;

#define BETA     0.9f
#define THRESH   1.0f
#define BATCH    256
#define T_STEPS  4096
#define DD       64
#define HH       192
#define NC       10
#define MT       16       // batch rows per block (WMMA M)
#define TCH      128      // timesteps staged per LDS chunk
#define NTHREADS 256      // 8 wave32

// LDS layout (float offsets). SSTRIDE=68 keeps state rows on distinct banks.
#define SSTRIDE  68
#define OFF_S0   0
#define OFF_S1   (OFF_S0 + MT * SSTRIDE)          // 1088
#define OFF_X    (OFF_S1 + MT * SSTRIDE)          // 2176
#define OFF_BV   (OFF_X + MT * TCH)               // 4224
#define OFF_EB   (OFF_BV + 64)                    // 4288
#define OFF_AT   (OFF_EB + HH)                    // 4480 : At = A^T, 64x64 (setup)
#define OFF_RATE OFF_AT                           // reuse At region for rate (3072)
#define LDS_FLOATS (OFF_AT + DD * DD)             // 8576 floats (~34 KB)

__global__ __launch_bounds__(NTHREADS)
void lsnn_fused_kernel(const float* __restrict__ x,
                       const float* __restrict__ enc_w,
                       const float* __restrict__ enc_b,
                       const float* __restrict__ ro_w,
                       const float* __restrict__ ro_b,
                       const float* __restrict__ A,
                       const float* __restrict__ Bv,
                       float* __restrict__ out)
{
    __shared__ float lds[LDS_FLOATS];
    const int tid  = threadIdx.x;
    const int wave = tid >> 5;
    const int lane = tid & 31;
    const int lo   = lane & 15;      // N column / A row within tile
    const int hi   = lane >> 4;      // half-wave select (K / M split)
    const int b0   = blockIdx.x * MT;

    // ---- one-time staging: At = A^T into LDS, B vector, enc bias ----------
    for (int i = tid; i < DD * DD; i += NTHREADS) {
        int k = i >> 6, n = i & 63;
        lds[OFF_AT + i] = A[n * DD + k];                // At[k][n] = A[n][k]
    }
    if (tid < DD) lds[OFF_BV + tid] = Bv[tid];
    if (tid < HH) lds[OFF_EB + tid] = enc_b[tid];
    for (int i = tid; i < 2 * MT * SSTRIDE; i += NTHREADS) lds[OFF_S0 + i] = 0.0f;
    __syncthreads();

    // ---- job assignment: 4 state tiles + 12 hidden tiles over 8 waves -----
    const bool job0_state = (wave < 4);
    const int  n0A = job0_state ? 16 * wave : 32 * (wave - 4);          // job0 col base
    const int  h0B = job0_state ? (128 + 16 * wave) : (32 * (wave - 4) + 16);

    // ---- preload constant B-operand fragments into registers --------------
    // B layout (16x16x4 f32): VGPR0 = row K (lanes0-15) / K+2 (lanes16-31), VGPR1 = K+1 / K+3
    v2f fB0[16], fB1[16];
    float cvec0, cvec1, cb0, cb1;
    if (job0_state) {
        const int n = n0A + lo;
        #pragma unroll
        for (int kb = 0; kb < 16; ++kb) {
            const int kk = 4 * kb + 2 * hi;
            fB0[kb].x = lds[OFF_AT + kk * DD + n];
            fB0[kb].y = lds[OFF_AT + (kk + 1) * DD + n];
        }
        cvec0 = lds[OFF_BV + n];                         // rank-1: u_t * B^T
        cb0   = 0.0f;
    } else {
        const int h = n0A + lo;
        #pragma unroll
        for (int kb = 0; kb < 16; ++kb) {
            const int kk = 4 * kb + 2 * hi;
            float s0 = 0.0f, s1 = 0.0f;
            for (int j = 0; j < DD; ++j) {               // W2 = At @ enc_w^T  (fused)
                const float w = enc_w[h * DD + j];
                s0 += lds[OFF_AT + kk * DD + j] * w;
                s1 += lds[OFF_AT + (kk + 1) * DD + j] * w;
            }
            fB0[kb].x = s0; fB0[kb].y = s1;
        }
        float b2 = 0.0f;                                 // b2 = B^T @ enc_w^T
        for (int j = 0; j < DD; ++j) b2 += lds[OFF_BV + j] * enc_w[h * DD + j];
        cvec0 = b2;
        cb0   = lds[OFF_EB + h];
    }
    {
        const int h = h0B + lo;                          // job1 is always a hidden tile
        #pragma unroll
        for (int kb = 0; kb < 16; ++kb) {
            const int kk = 4 * kb + 2 * hi;
            float s0 = 0.0f, s1 = 0.0f;
            for (int j = 0; j < DD; ++j) {
                const float w = enc_w[h * DD + j];
                s0 += lds[OFF_AT + kk * DD + j] * w;
                s1 += lds[OFF_AT + (kk + 1) * DD + j] * w;
            }
            fB1[kb].x = s0; fB1[kb].y = s1;
        }
        float b2 = 0.0f;
        for (int j = 0; j < DD; ++j) b2 += lds[OFF_BV + j] * enc_w[h * DD + j];
        cvec1 = b2;
        cb1   = lds[OFF_EB + h];
    }

    v8f mem0 = {}, mem1 = {}, ss0 = {}, ss1 = {};
    float* __restrict__ spk_out = out + BATCH * NC;

    // ======================= sequential scan over T ========================
    for (int t0 = 0; t0 < T_STEPS; t0 += TCH) {
        // stage input chunk: lds_x[m][tt], coalesced 128-float rows
        for (int i = tid; i < MT * TCH; i += NTHREADS) {
            const int m = i >> 7, tt = i & (TCH - 1);
            lds[OFF_X + i] = x[(size_t)(b0 + m) * T_STEPS + t0 + tt];
        }
        __syncthreads();

        for (int tt = 0; tt < TCH; ++tt) {
            const int t    = t0 + tt;
            const int rbuf = (t & 1) ? OFF_S1 : OFF_S0;  // s_{t-1}
            const int wbuf = (t & 1) ? OFF_S0 : OFF_S1;  // s_t
            const float* sp = &lds[rbuf + lo * SSTRIDE + 2 * hi];

            // u_t for accumulator init (C-layout row M = v + 8*hi)
            float um[8];
            #pragma unroll
            for (int v = 0; v < 8; ++v) um[v] = lds[OFF_X + (v + 8 * hi) * TCH + tt];

            // 4 independent accumulator chains (2 tiles x 2 partials)
            v8f acc0, acc1, acc0b = {}, acc1b = {};
            #pragma unroll
            for (int v = 0; v < 8; ++v) {
                acc0[v] = um[v] * cvec0 + cb0;
                acc1[v] = um[v] * cvec1 + cb1;
            }
            // interleaved WMMA stream: dep distance 4 keeps the XDL pipe full
            #pragma unroll
            for (int kb = 0; kb < 8; ++kb) {
                const v2f aA = *(const v2f*)(sp + 8 * kb);       // K-block 2kb
                const v2f aB = *(const v2f*)(sp + 8 * kb + 4);   // K-block 2kb+1
                acc0  = __builtin_amdgcn_wmma_f32_16x16x4_f32(
                          false, aA, false, fB0[2 * kb],     (short)0, acc0,  false, false);
                acc1  = __builtin_amdgcn_wmma_f32_16x16x4_f32(
                          false, aA, false, fB1[2 * kb],     (short)0, acc1,  false, false);
                acc0b = __builtin_amdgcn_wmma_f32_16x16x4_f32(
                          false, aB, false, fB0[2 * kb + 1], (short)0, acc0b, false, false);
                acc1b = __builtin_amdgcn_wmma_f32_16x16x4_f32(
                          false, aB, false, fB1[2 * kb + 1], (short)0, acc1b, false, false);
            }
            #pragma unroll
            for (int v = 0; v < 8; ++v) { acc0[v] += acc0b[v]; acc1[v] += acc1b[v]; }

            if (job0_state) {                            // write s_t tile to LDS
                float* dp = &lds[wbuf + hi * 8 * SSTRIDE + n0A + lo];
                #pragma unroll
                for (int v = 0; v < 8; ++v) dp[v * SSTRIDE] = acc0[v];
            }

            // drain own LDS traffic, release barrier early; LIF + spike
            // stores below execute in the barrier's shadow.
            __asm__ volatile("s_wait_dscnt 0\n\ts_barrier_signal -1" ::: "memory");

            if (!job0_state) {                           // LIF update + spike emit
                #pragma unroll
                for (int v = 0; v < 8; ++v) {
                    const float m_ = BETA * mem0[v] + acc0[v];
                    const float s  = (m_ > THRESH) ? 1.0f : 0.0f;
                    spk_out[((size_t)(b0 + v + 8 * hi) * T_STEPS + t) * HH + n0A + lo] = s;
                    ss0[v]  += s;
                    mem0[v]  = m_ * (1.0f - s);
                }
            }
            #pragma unroll
            for (int v = 0; v < 8; ++v) {
                const float m_ = BETA * mem1[v] + acc1[v];
                const float s  = (m_ > THRESH) ? 1.0f : 0.0f;
                spk_out[((size_t)(b0 + v + 8 * hi) * T_STEPS + t) * HH + h0B + lo] = s;
                ss1[v]  += s;
                mem1[v]  = m_ * (1.0f - s);
            }

            __asm__ volatile("s_barrier_wait -1" ::: "memory");
        }
    }

    // ---- rate readout: rate = mean_t spk; out = rate @ ro_w^T + ro_b ------
    const float invT = 1.0f / (float)T_STEPS;
    if (!job0_state) {
        #pragma unroll
        for (int v = 0; v < 8; ++v)
            lds[OFF_RATE + (v + 8 * hi) * HH + n0A + lo] = ss0[v] * invT;
    }
    #pragma unroll
    for (int v = 0; v < 8; ++v)
        lds[OFF_RATE + (v + 8 * hi) * HH + h0B + lo] = ss1[v] * invT;
    __syncthreads();

    for (int i = tid; i < MT * NC; i += NTHREADS) {
        const int m = i / NC, c = i % NC;
        float o = ro_b[c];
        for (int h = 0; h < HH; ++h)
            o += lds[OFF_RATE + m * HH + h] * ro_w[c * HH + h];
        out[(b0 + m) * NC + c] = o;
    }
}

extern "C" void kernel_launch(void* const* d_in, const int* in_sizes, int n_in,
                              void* d_out, int out_size, void* d_ws, size_t ws_size,
                              hipStream_t stream) {
    const float* x     = (const float*)d_in[0];
    const float* enc_w = (const float*)d_in[1];
    const float* enc_b = (const float*)d_in[2];
    const float* ro_w  = (const float*)d_in[3];
    const float* ro_b  = (const float*)d_in[4];
    const float* A     = (const float*)d_in[5];
    const float* Bv    = (const float*)d_in[6];
    float* out = (float*)d_out;

    lsnn_fused_kernel<<<BATCH / MT, NTHREADS, 0, stream>>>(
        x, enc_w, enc_b, ro_w, ro_b, A, Bv, out);
}